// FAGCN_75204877353214
// MI455X (gfx1250) — compile-verified
//
#include <hip/hip_runtime.h>

#define N_NODES 50000
#define N_EDGES 600000
#define IN_DIM  512
#define HIDDEN  128
#define OUT_DIM 64
#define N_LAYERS 3
#define EPS_F   0.1f

typedef __attribute__((ext_vector_type(2))) float v2f;
typedef __attribute__((ext_vector_type(4))) float v4f;
typedef __attribute__((ext_vector_type(8))) float v8f;

// ---------------------------------------------------------------------------
// Degree accumulation (hardware f32 atomics)
// ---------------------------------------------------------------------------
__global__ void k_deg(const int* __restrict__ dst, float* __restrict__ deg) {
    int e = blockIdx.x * blockDim.x + threadIdx.x;
    if (e < N_EDGES) unsafeAtomicAdd(&deg[dst[e]], 1.0f);
}

__global__ void k_dinv(float* __restrict__ deg) {
    int i = blockIdx.x * blockDim.x + threadIdx.x;
    if (i < N_NODES) {
        float d = deg[i];
        deg[i] = (d > 0.0f) ? rsqrtf(d) : 0.0f;
    }
}

__global__ void k_norm(const int* __restrict__ src, const int* __restrict__ dst,
                       const float* __restrict__ dinv, float* __restrict__ nrm) {
    int e = blockIdx.x * blockDim.x + threadIdx.x;
    if (e < N_EDGES) nrm[e] = dinv[src[e]] * dinv[dst[e]];
}

// ---------------------------------------------------------------------------
// Shared WMMA fp32 GEMM tile body:  out[16 x (NT*16)] = A[16 x K] * W[NT*16 x K]^T
// V_WMMA_F32_16X16X4_F32 layouts:
//   A: lane m = lane&15, K-pair = (lane>>4)*2 + {0,1}   (contiguous float2)
//   B: lane n = lane&15, same K-pair; B = W^T -> contiguous float2 of W row n
//   C: VGPR r -> row = (lane>>4)*8 + r, col = lane&15
// All B fragments for a k-step are loaded into an array FIRST so the loads
// issue as one clause (distinct regs, many outstanding) before the WMMA batch.
// ---------------------------------------------------------------------------
template <int NT, int K>
__device__ __forceinline__ void wmma_tile(const float* __restrict__ arow,
                                          const float* __restrict__ wbase,
                                          v8f (&acc)[NT]) {
#pragma unroll 4
    for (int kk = 0; kk < K / 4; ++kk) {
        v2f a = *(const v2f*)(arow + kk * 4);
        v2f b[NT];
#pragma unroll
        for (int tn = 0; tn < NT; ++tn)
            b[tn] = *(const v2f*)(wbase + (size_t)tn * 16 * K + kk * 4);
#pragma unroll
        for (int tn = 0; tn < NT; ++tn)
            acc[tn] = __builtin_amdgcn_wmma_f32_16x16x4_f32(
                false, a, false, b[tn], (short)0, acc[tn], false, false);
    }
}

// GEMM1: h = relu(x @ t1_w^T + b1)   [50000,512] x [128,512]^T -> [50000,128]
__global__ void k_gemm1_relu(const float* __restrict__ x,
                             const float* __restrict__ w,
                             const float* __restrict__ bias,
                             float* __restrict__ out) {
    const int tm   = blockIdx.x;          // 0..3124
    const int lane = threadIdx.x;         // wave32
    const int nlo  = lane & 15;
    const int hi   = lane >> 4;

    const float* arow  = x + (size_t)(tm * 16 + nlo) * IN_DIM + hi * 2;
    const float* wbase = w + (size_t)nlo * IN_DIM + hi * 2;

    v8f acc[8] = {};
    wmma_tile<8, IN_DIM>(arow, wbase, acc);

    const int m0 = tm * 16 + hi * 8;
#pragma unroll
    for (int tn = 0; tn < 8; ++tn) {
        const int col = tn * 16 + nlo;
        const float bv = bias[col];
#pragma unroll
        for (int r = 0; r < 8; ++r) {
            float v = acc[tn][r] + bv;
            out[(size_t)(m0 + r) * HIDDEN + col] = v > 0.0f ? v : 0.0f;
        }
    }
}

// GEMM2: emb = h @ t2_w^T + b2   [50000,128] x [64,128]^T -> [50000,64]
__global__ void k_gemm2(const float* __restrict__ h,
                        const float* __restrict__ w,
                        const float* __restrict__ bias,
                        float* __restrict__ emb) {
    const int tm   = blockIdx.x;
    const int lane = threadIdx.x;
    const int nlo  = lane & 15;
    const int hi   = lane >> 4;

    const float* arow  = h + (size_t)(tm * 16 + nlo) * HIDDEN + hi * 2;
    const float* wbase = w + (size_t)nlo * HIDDEN + hi * 2;

    v8f acc[4] = {};
    wmma_tile<4, HIDDEN>(arow, wbase, acc);

    const int m0 = tm * 16 + hi * 8;
#pragma unroll
    for (int tn = 0; tn < 4; ++tn) {
        const int col = tn * 16 + nlo;
        const float bv = bias[col];
#pragma unroll
        for (int r = 0; r < 8; ++r)
            emb[(size_t)(m0 + r) * OUT_DIM + col] = acc[tn][r] + bv;
    }
}

// ---------------------------------------------------------------------------
// Per-node attention dots: al = h . att_l, ar = h . att_r  (wave per node)
// ---------------------------------------------------------------------------
__global__ void k_att(const float* __restrict__ h,
                      const float* __restrict__ attl,
                      const float* __restrict__ attr,
                      float* __restrict__ al, float* __restrict__ ar) {
    int gid  = blockIdx.x * blockDim.x + threadIdx.x;
    int node = gid >> 5;
    int lane = threadIdx.x & 31;
    if (node >= N_NODES) return;
    const float* row = h + (size_t)node * HIDDEN;
    float sl = 0.0f, sr = 0.0f;
#pragma unroll
    for (int j = 0; j < 4; ++j) {
        int d = lane + 32 * j;
        float v = row[d];
        sl += v * attl[d];
        sr += v * attr[d];
    }
#pragma unroll
    for (int off = 16; off > 0; off >>= 1) {
        sl += __shfl_xor(sl, off, 32);
        sr += __shfl_xor(sr, off, 32);
    }
    if (lane == 0) { al[node] = sl; ar[node] = sr; }
}

// ---------------------------------------------------------------------------
// h_next = EPS * raw  (scatter destinations pre-initialized -> fuses +EPS*raw)
// ---------------------------------------------------------------------------
__global__ void k_scale_init(const float* __restrict__ raw, float* __restrict__ out) {
    int i = blockIdx.x * blockDim.x + threadIdx.x;
    if (i < N_NODES * HIDDEN / 4) {
        v4f v = ((const v4f*)raw)[i];
        ((v4f*)out)[i] = v * EPS_F;
    }
}

// ---------------------------------------------------------------------------
// Edge scatter: one wave per edge; 32 lanes x float4 covers 128 features.
// coef = tanh(al[src]+ar[dst]) * norm;  out[dst] += h[src] * coef (hw f32 atomics)
// ---------------------------------------------------------------------------
__global__ void k_scatter(const float* __restrict__ h,
                          const int* __restrict__ src, const int* __restrict__ dst,
                          const float* __restrict__ al, const float* __restrict__ ar,
                          const float* __restrict__ nrm,
                          float* __restrict__ out) {
    int gid  = blockIdx.x * blockDim.x + threadIdx.x;
    int e    = gid >> 5;
    int lane = threadIdx.x & 31;
    if (e >= N_EDGES) return;
    int s = src[e], d = dst[e];
    float coef = tanhf(al[s] + ar[d]) * nrm[e];
    v4f hv = *(const v4f*)(h + (size_t)s * HIDDEN + lane * 4);
    float* o = out + (size_t)d * HIDDEN + lane * 4;
    unsafeAtomicAdd(o + 0, hv.x * coef);
    unsafeAtomicAdd(o + 1, hv.y * coef);
    unsafeAtomicAdd(o + 2, hv.z * coef);
    unsafeAtomicAdd(o + 3, hv.w * coef);
}

// ---------------------------------------------------------------------------
// Row log-softmax over 64 cols (wave per node, 2 cols per lane)
// ---------------------------------------------------------------------------
__global__ void k_logsoftmax(const float* __restrict__ emb, float* __restrict__ out) {
    int gid  = blockIdx.x * blockDim.x + threadIdx.x;
    int node = gid >> 5;
    int lane = threadIdx.x & 31;
    if (node >= N_NODES) return;
    const float* row = emb + (size_t)node * OUT_DIM;
    float v0 = row[lane], v1 = row[lane + 32];
    float m = fmaxf(v0, v1);
#pragma unroll
    for (int off = 16; off > 0; off >>= 1)
        m = fmaxf(m, __shfl_xor(m, off, 32));
    float s = __expf(v0 - m) + __expf(v1 - m);
#pragma unroll
    for (int off = 16; off > 0; off >>= 1)
        s += __shfl_xor(s, off, 32);
    float lse = m + __logf(s);
    float* o = out + (size_t)node * OUT_DIM;
    o[lane]      = v0 - lse;
    o[lane + 32] = v1 - lse;
}

// ---------------------------------------------------------------------------
extern "C" void kernel_launch(void* const* d_in, const int* in_sizes, int n_in,
                              void* d_out, int out_size, void* d_ws, size_t ws_size,
                              hipStream_t stream) {
    const float* x    = (const float*)d_in[0];
    const int*   ei   = (const int*)d_in[1];
    const float* t1w  = (const float*)d_in[2];
    const float* t1b  = (const float*)d_in[3];
    const float* t2w  = (const float*)d_in[4];
    const float* t2b  = (const float*)d_in[5];
    const float* attl = (const float*)d_in[6];  // [3,128]
    const float* attr = (const float*)d_in[7];  // [3,128]
    const int* src = ei;
    const int* dst = ei + N_EDGES;

    float* out = (float*)d_out;                       // [N,64] log_softmax
    float* emb = out + (size_t)N_NODES * OUT_DIM;     // [N,64] emb (2nd tuple output)

    // workspace layout (floats, 64-aligned element counts)
    float* w    = (float*)d_ws;
    float* dinv = w;                                  // 50048
    float* nrm  = w + 50048;                          // 600000
    float* al   = w + 650048;                         // 50048
    float* ar   = w + 700096;                         // 50048
    float* raw  = w + 750144;                         // 6,400,000
    float* hA   = raw + (size_t)N_NODES * HIDDEN;     // 6,400,000
    float* hB   = hA  + (size_t)N_NODES * HIDDEN;     // 6,400,000

    hipMemsetAsync(dinv, 0, N_NODES * sizeof(float), stream);
    k_deg <<<(N_EDGES + 255) / 256, 256, 0, stream>>>(dst, dinv);
    k_dinv<<<(N_NODES + 255) / 256, 256, 0, stream>>>(dinv);
    k_norm<<<(N_EDGES + 255) / 256, 256, 0, stream>>>(src, dst, dinv, nrm);

    k_gemm1_relu<<<N_NODES / 16, 32, 0, stream>>>(x, t1w, t1b, raw);

    const float* hcur = raw;
    float* pingpong[2] = { hA, hB };
    for (int L = 0; L < N_LAYERS; ++L) {
        float* hn = pingpong[L & 1];
        k_att<<<(N_NODES * 32 + 255) / 256, 256, 0, stream>>>(
            hcur, attl + L * HIDDEN, attr + L * HIDDEN, al, ar);
        k_scale_init<<<(N_NODES * HIDDEN / 4 + 255) / 256, 256, 0, stream>>>(raw, hn);
        k_scatter<<<(N_EDGES * 32 + 255) / 256, 256, 0, stream>>>(
            hcur, src, dst, al, ar, nrm, hn);
        hcur = hn;
    }

    k_gemm2<<<N_NODES / 16, 32, 0, stream>>>(hcur, t2w, t2b, emb);
    k_logsoftmax<<<(N_NODES * 32 + 255) / 256, 256, 0, stream>>>(emb, out);
}